// EMLTree1DLinear_9191230013489
// MI455X (gfx1250) — compile-verified
//
#include <hip/hip_runtime.h>
#include <hip/hip_bf16.h>

#define DEPTH       10
#define N_LEAVES    1024          // 2^DEPTH
#define N_PAIRS     512
#define N_INTERNAL  1023
#define BLOCK       256

// ---- async global->LDS staging (probe-verified builtin signature) ----
#if defined(__HIP_DEVICE_COMPILE__) && defined(__has_builtin)
#if __has_builtin(__builtin_amdgcn_global_load_async_to_lds_b32) && \
    __has_builtin(__builtin_amdgcn_s_wait_asynccnt)
#define EML_ASYNC_STAGE 1
#endif
#endif
#ifndef EML_ASYNC_STAGE
#define EML_ASYNC_STAGE 0
#endif

typedef __attribute__((address_space(1))) int* gint_ptr;
typedef __attribute__((address_space(3))) int* lint_ptr;

__device__ __forceinline__ void stage_b32(const float* __restrict__ gsrc,
                                          float* __restrict__ ldst) {
#if EML_ASYNC_STAGE
  __builtin_amdgcn_global_load_async_to_lds_b32(
      (gint_ptr)gsrc, (lint_ptr)ldst, /*offset=*/0, /*cpol=*/0);
#else
  *ldst = *gsrc;
#endif
}

__device__ __forceinline__ void stage_wait() {
#if EML_ASYNC_STAGE
  __builtin_amdgcn_s_wait_asynccnt(0);
#endif
}

// ---- math helpers (raw gfx1250 transcendental ops) ----
__device__ __forceinline__ float sanit(float v) {
  v = (v != v) ? 0.0f : v;                       // NaN -> 0
  return fminf(fmaxf(v, -1.0e6f), 1.0e6f);       // clamps +-inf too
}

// atan2 via poly + v_rcp_f32 (principal branch; z==0 -> NaN, sanitized later)
__device__ __forceinline__ float fast_atan2(float yy, float xx) {
  float ax = fabsf(xx), ay = fabsf(yy);
  float mn = fminf(ax, ay), mx = fmaxf(ax, ay);
  float a  = mn * __builtin_amdgcn_rcpf(mx);     // in [0,1]
  float s  = a * a;
  float r  = 0.0208351f;
  r = __builtin_fmaf(r, s, -0.0851330f);
  r = __builtin_fmaf(r, s,  0.1801410f);
  r = __builtin_fmaf(r, s, -0.3302995f);
  r = __builtin_fmaf(r, s,  0.9998660f);
  r = r * a;
  r = (ay > ax)   ? 1.57079633f - r : r;
  r = (xx < 0.0f) ? 3.14159265f - r : r;
  r = (yy < 0.0f) ? -r : r;
  return r;
}

// cexp(wr + i*wi) with sanitize (v_exp/v_sin/v_cos take log2/cycle units)
__device__ __forceinline__ void cexp_sanit(float wr, float wi,
                                           float& ore, float& oim) {
  float m = __builtin_amdgcn_exp2f(wr * 1.44269504f);
  float c = __builtin_amdgcn_cosf(wi * 0.15915494f);
  float s = __builtin_amdgcn_sinf(wi * 0.15915494f);
  ore = sanit(m * c);
  oim = sanit(m * s);
}

// generic internal node: v (right, in/out) combined with (lre,lim) as left child
__device__ __forceinline__ void eml_combine(float& vre, float& vim,
                                            float lre, float lim,
                                            const float* __restrict__ g, float x) {
  const float rre = vre, rim = vim;
  // left_in = g0 + g1*x + g2*L ; right_in = g3 + g4*x + g5*R   (g real)
  float li_re = __builtin_fmaf(g[2], lre, __builtin_fmaf(g[1], x, g[0]));
  float li_im = g[2] * lim;
  float ri_re = __builtin_fmaf(g[5], rre, __builtin_fmaf(g[4], x, g[3]));
  float ri_im = g[5] * rim;
  // complex log: 0.5*ln(re^2+im^2) = 0.5*ln2*log2(.) ; arg = atan2
  float lmag = 0.34657359f *
      __builtin_amdgcn_logf(__builtin_fmaf(li_re, li_re, li_im * li_im));
  float larg = fast_atan2(li_im, li_re);
  float rmag = 0.34657359f *
      __builtin_amdgcn_logf(__builtin_fmaf(ri_re, ri_re, ri_im * ri_im));
  float rarg = fast_atan2(ri_im, ri_re);
  // w = log(u)*log(v)
  float wr = lmag * rmag - larg * rarg;
  float wi = lmag * rarg + larg * rmag;
  cexp_sanit(wr, wi, vre, vim);
}

// bottom-level node: both children purely real -> log degenerates,
// arg in {0, pi}; no atan2, no imaginary FMAs
__device__ __forceinline__ void eml_combine_real(float l, float r,
                                                 const float* __restrict__ g,
                                                 float x,
                                                 float& vre, float& vim) {
  float li = __builtin_fmaf(g[2], l, __builtin_fmaf(g[1], x, g[0]));
  float ri = __builtin_fmaf(g[5], r, __builtin_fmaf(g[4], x, g[3]));
  // ln|v| = 0.5*ln(v^2) = 0.5*ln2*log2(v^2)  (no fabs needed)
  float lmag = 0.34657359f * __builtin_amdgcn_logf(li * li);
  float rmag = 0.34657359f * __builtin_amdgcn_logf(ri * ri);
  float larg = (li < 0.0f) ? 3.14159265f : 0.0f;
  float rarg = (ri < 0.0f) ? 3.14159265f : 0.0f;
  float wr = lmag * rmag - larg * rarg;
  float wi = lmag * rarg + larg * rmag;
  cexp_sanit(wr, wi, vre, vim);
}

__global__ __launch_bounds__(BLOCK)
void eml_tree_kernel(const float* __restrict__ x_in,
                     const float* __restrict__ leaf,   // [1024][2]
                     const float* __restrict__ gate,   // [1023][2][3]
                     float* __restrict__ out,          // [batch][2] complex64
                     int batch) {
  __shared__ float s_leaf[2 * N_LEAVES];       // 8 KB  (pair p2 at [4*p2..4*p2+3])
  __shared__ float s_gate[8 * N_INTERNAL + 8]; // padded to 32B/node, ~32 KB

  // stage coefficient tables into LDS (wave-uniform broadcast reads later)
  for (int i = threadIdx.x; i < 2 * N_LEAVES; i += BLOCK)
    stage_b32(&leaf[i], &s_leaf[i]);
  for (int node = threadIdx.x; node < N_INTERNAL; node += BLOCK) {
#pragma unroll
    for (int k = 0; k < 6; ++k)
      stage_b32(&gate[node * 6 + k], &s_gate[node * 8 + k]);
  }
  stage_wait();
  __syncthreads();

  const int tid = blockIdx.x * BLOCK + threadIdx.x;
  if (tid >= batch) return;
  const float x = x_in[tid];

  // binary-counter carries: carry[t] = pending left value at level t (t=1..9)
  float cre[DEPTH], cim[DEPTH];
  float vre = 0.0f, vim = 0.0f;

#pragma unroll 1
  for (int p2 = 0; p2 < N_PAIRS; ++p2) {
    // two leaves of this pair: one ds_load_b128 (a0,b0,a1,b1)
    float a0 = s_leaf[4 * p2 + 0], b0 = s_leaf[4 * p2 + 1];
    float a1 = s_leaf[4 * p2 + 2], b1 = s_leaf[4 * p2 + 3];
    float l = __builtin_fmaf(b0, x, a0);
    float r = __builtin_fmaf(b1, x, a1);
    // bottom level (tag 0): gate index == pair index, real-specialized
    eml_combine_real(l, r, &s_gate[p2 * 8], x, vre, vim);
    // propagate carries up the tree (uniform control flow: p2 is uniform)
#pragma unroll
    for (int t = 1; t < DEPTH; ++t) {
      if (((p2 >> (t - 1)) & 1) == 0) { cre[t] = vre; cim[t] = vim; break; }
      int gidx = (N_LEAVES - (N_LEAVES >> t)) + (p2 >> t);
      eml_combine(vre, vim, cre[t], cim[t], &s_gate[gidx * 8], x);
    }
  }
  // p2 = 511 has all bits set -> v holds the root
  out[2 * tid + 0] = vre;
  out[2 * tid + 1] = vim;
}

extern "C" void kernel_launch(void* const* d_in, const int* in_sizes, int n_in,
                              void* d_out, int out_size, void* d_ws, size_t ws_size,
                              hipStream_t stream) {
  const float* x    = (const float*)d_in[0];
  const float* leaf = (const float*)d_in[1];
  const float* gate = (const float*)d_in[2];
  float* out        = (float*)d_out;
  int batch  = in_sizes[0];
  int blocks = (batch + BLOCK - 1) / BLOCK;
  hipLaunchKernelGGL(eml_tree_kernel, dim3(blocks), dim3(BLOCK), 0, stream,
                     x, leaf, gate, out, batch);
}